// FRMIL_65644280152857
// MI455X (gfx1250) — compile-verified
//
#include <hip/hip_runtime.h>
#include <hip/hip_bf16.h>

#define BB 8
#define DD 1024
#define NN 4096
#define NX 4097
#define HH 8
#define DHD 128
#define APAD 4100   // n padded to multiple of 4, zero-filled

typedef __attribute__((ext_vector_type(2))) float v2f;
typedef __attribute__((ext_vector_type(8))) float v8f;

// ---------------- Kernel 1: argmax over sigmoid(x·w) (monotonic -> argmax of dot), extract Q row
__global__ void k_argmax(const float* __restrict__ inputs, const float* __restrict__ W_enc,
                         float* __restrict__ Qout, int* __restrict__ idxout) {
    int b = blockIdx.x;
    __shared__ float sW[DD];
    __shared__ float swval[8];
    __shared__ int   swidx[8];
    __shared__ int   sidx;
    for (int i = threadIdx.x; i < DD; i += blockDim.x) sW[i] = W_enc[i];
    __syncthreads();
    int wave = threadIdx.x >> 5, lane = threadIdx.x & 31;
    const float* inb = inputs + (size_t)b * NN * DD;
    float best = -3.4e38f; int bidx = 0;
    for (int n = wave; n < NN; n += 8) {
        const float* row = inb + (size_t)n * DD;
        float p = 0.f;
        for (int d = lane; d < DD; d += 32) p += row[d] * sW[d];
        for (int off = 16; off > 0; off >>= 1) p += __shfl_xor(p, off, 32);
        if (p > best) { best = p; bidx = n; }   // ascending n per wave -> first max kept
    }
    if (lane == 0) { swval[wave] = best; swidx[wave] = bidx; }
    __syncthreads();
    if (threadIdx.x == 0) {
        float bb = swval[0]; int bi = swidx[0];
        for (int w = 1; w < 8; w++)
            if (swval[w] > bb || (swval[w] == bb && swidx[w] < bi)) { bb = swval[w]; bi = swidx[w]; }
        sidx = bi; idxout[b] = bi;
    }
    __syncthreads();
    const float* qrow = inb + (size_t)sidx * DD;
    for (int d = threadIdx.x; d < DD; d += blockDim.x) Qout[b * DD + d] = qrow[d];
}

// ---------------- Kernel 2a: cls token -> x row 0
__global__ void k_cls(const float* __restrict__ cls_token, float* __restrict__ X) {
    int b = blockIdx.x;
    for (int d = threadIdx.x; d < DD; d += blockDim.x)
        X[(size_t)b * NX * DD + d] = cls_token[d];
}

// ---------------- Kernel 2b: depthwise 3x3 conv over relu(inputs - Q), +bias +residual -> x rows 1..4096
#define DC 64
__global__ void k_conv(const float* __restrict__ inputs, const float* __restrict__ Qv,
                       const float* __restrict__ conv_w, const float* __restrict__ conv_b,
                       float* __restrict__ X) {
    int b = blockIdx.x;
    int r = blockIdx.y;          // spatial row 0..63
    int d0 = blockIdx.z * DC;    // channel chunk
    __shared__ float tile[3][64][DC];   // 49152 B
    __shared__ float sq[DC];
    const float* inb = inputs + (size_t)b * NN * DD;
    const float* qb  = Qv + (size_t)b * DD;
    for (int i = threadIdx.x; i < DC; i += blockDim.x) sq[i] = qb[d0 + i];
    __syncthreads();
    for (int rr = 0; rr < 3; rr++) {
        int rs = r + rr - 1;
        for (int i = threadIdx.x; i < 64 * DC; i += blockDim.x) {
            int c = i / DC, dd = i % DC;
            float v = 0.f;
            if (rs >= 0 && rs < 64) {
                float t = inb[(size_t)(rs * 64 + c) * DD + d0 + dd] - sq[dd];
                v = t > 0.f ? t : 0.f;
            }
            tile[rr][c][dd] = v;
        }
    }
    __syncthreads();
    for (int i = threadIdx.x; i < 64 * DC; i += blockDim.x) {
        int c = i / DC, dd = i % DC;
        int d = d0 + dd;
        const float* w = conv_w + (size_t)d * 9;
        float acc = conv_b[d];
        #pragma unroll
        for (int ii = 0; ii < 3; ii++)
            #pragma unroll
            for (int jj = 0; jj < 3; jj++) {
                int cc = c + jj - 1;
                if (cc >= 0 && cc < 64) acc += w[ii * 3 + jj] * tile[ii][cc][dd];
            }
        acc += tile[1][c][dd];   // residual
        X[(size_t)b * NX * DD + (size_t)(1 + r * 64 + c) * DD + d] = acc;
    }
}

// ---------------- GEMV: out[b,i] = bias[i] + vec[b]·W[i,:]; optional out = vec + relu(.)
__global__ void k_gemv_rowdot(const float* __restrict__ vec, const float* __restrict__ W,
                              const float* __restrict__ bias, float* __restrict__ out,
                              int relu_residual) {
    int b = blockIdx.x;
    int wave = threadIdx.x >> 5, lane = threadIdx.x & 31;
    int i = blockIdx.y * 8 + wave;
    const float* v = vec + (size_t)b * DD;
    const float* w = W + (size_t)i * DD;
    float p = 0.f;
    for (int d = lane; d < DD; d += 32) p += v[d] * w[d];
    for (int off = 16; off > 0; off >>= 1) p += __shfl_xor(p, off, 32);
    if (lane == 0) {
        float r = p + bias[i];
        if (relu_residual) r = v[i] + (r > 0.f ? r : 0.f);
        out[b * DD + i] = r;
    }
}

// ---------------- G[b,h,d] = sum_j Qp[b,h*128+j] * Wk[h*128+j, d]   (rows h>=8 pre-zeroed)
__global__ void k_g(const float* __restrict__ Qp, const float* __restrict__ Wk,
                    float* __restrict__ G) {
    int b = blockIdx.x, h = blockIdx.y;
    int d = blockIdx.z * 256 + threadIdx.x;
    const float* q = Qp + (size_t)b * DD + h * DHD;
    float acc = 0.f;
    for (int j = 0; j < DHD; j++) acc += q[j] * Wk[(size_t)(h * DHD + j) * DD + d];
    G[((size_t)b * 16 + h) * DD + d] = acc;
}

// ---------------- logits_raw[b,n,h] = x[b,n,:]·G[b,h,:]  via V_WMMA_F32_16X16X4_F32
__global__ void k_logits(const float* __restrict__ X, const float* __restrict__ G,
                         float* __restrict__ Lraw) {
    int b = blockIdx.x;
    int wave = threadIdx.x >> 5, lane = threadIdx.x & 31;
    int tile = blockIdx.y * 4 + wave;     // 257 tiles of 16 rows
    if (tile >= 257) return;
    int n0 = tile * 16;
    const float* xb = X + (size_t)b * NX * DD;
    const float* gb = G + (size_t)b * 16 * DD;
    int l15 = lane & 15;
    int khalf = (lane >> 4) * 2;          // 0 or 2
    int nrow = n0 + l15; if (nrow > NN) nrow = NN;   // clamp; stores masked below
    const float* xr = xb + (size_t)nrow * DD;
    const float* gr = gb + (size_t)l15 * DD;
    v8f acc = {};
    for (int k = 0; k < DD; k += 4) {
        v2f a, bm;
        a.x  = xr[k + khalf];     a.y  = xr[k + khalf + 1];   // A 16x4 layout
        bm.x = gr[k + khalf];     bm.y = gr[k + khalf + 1];   // B 4x16 layout
        acc = __builtin_amdgcn_wmma_f32_16x16x4_f32(false, a, false, bm,
                                                    (short)0, acc, false, false);
    }
    int mbase = (lane >> 4) * 8;
    #pragma unroll
    for (int v = 0; v < 8; v++) {
        int n = n0 + mbase + v;
        if (n < NX) Lraw[((size_t)b * NX + n) * 16 + l15] = acc[v];
    }
}

// ---------------- softmax over n of (raw/32); bk shift cancels. Writes padded A (B,16,4100)
__global__ void k_softmax(const float* __restrict__ Lraw, float* __restrict__ Apd) {
    int b = blockIdx.x, h = blockIdx.y;
    __shared__ float red[256];
    const float* lb = Lraw + (size_t)b * NX * 16 + h;
    float m = -3.4e38f;
    for (int n = threadIdx.x; n < NX; n += 256) m = fmaxf(m, lb[(size_t)n * 16]);
    red[threadIdx.x] = m; __syncthreads();
    for (int s = 128; s > 0; s >>= 1) {
        if (threadIdx.x < s) red[threadIdx.x] = fmaxf(red[threadIdx.x], red[threadIdx.x + s]);
        __syncthreads();
    }
    m = red[0]; __syncthreads();
    float* ab = Apd + ((size_t)b * 16 + h) * APAD;
    float s = 0.f;
    for (int n = threadIdx.x; n < NX; n += 256) {
        float e = expf((lb[(size_t)n * 16] - m) * 0.03125f);   // 1/sqrt(1024)
        ab[n] = e; s += e;
    }
    red[threadIdx.x] = s; __syncthreads();
    for (int st = 128; st > 0; st >>= 1) {
        if (threadIdx.x < st) red[threadIdx.x] += red[threadIdx.x + st];
        __syncthreads();
    }
    float inv = 1.f / red[0];
    for (int n = threadIdx.x; n < NX; n += 256) ab[n] *= inv;
}

// ---------------- u[b,h,:] = A[b,h,:] @ x[b]  via V_WMMA_F32_16X16X4_F32 (M=16 heads pad)
__global__ void k_u(const float* __restrict__ X, const float* __restrict__ Apd,
                    float* __restrict__ U) {
    int b = blockIdx.x;
    int wave = threadIdx.x >> 5, lane = threadIdx.x & 31;
    int d0 = (blockIdx.y * 8 + wave) * 16;   // 64 d-tiles
    const float* xb = X + (size_t)b * NX * DD;
    const float* ab = Apd + (size_t)b * 16 * APAD;
    int l15 = lane & 15;
    int khalf = (lane >> 4) * 2;
    const float* ar = ab + (size_t)l15 * APAD;
    v8f acc = {};
    for (int k = 0; k < APAD; k += 4) {
        v2f a, bm;
        a.x = ar[k + khalf]; a.y = ar[k + khalf + 1];       // A rows = heads, K = n
        int n1 = k + khalf;     if (n1 > NN) n1 = NN;       // A==0 in padding -> safe clamp
        int n2 = k + khalf + 1; if (n2 > NN) n2 = NN;
        bm.x = xb[(size_t)n1 * DD + d0 + l15];
        bm.y = xb[(size_t)n2 * DD + d0 + l15];
        acc = __builtin_amdgcn_wmma_f32_16x16x4_f32(false, a, false, bm,
                                                    (short)0, acc, false, false);
    }
    if (lane < 16) {                      // M=0..7 live in lanes 0-15 -> heads 0..7
        #pragma unroll
        for (int v = 0; v < 8; v++)
            U[((size_t)b * HH + v) * DD + d0 + lane] = acc[v];
    }
}

// ---------------- O1[b,i] = Qp[b,i] + bv[i] + u[b,i/128,:]·Wv[i,:]
__global__ void k_o(const float* __restrict__ U, const float* __restrict__ Wv,
                    const float* __restrict__ Qp, const float* __restrict__ bv,
                    float* __restrict__ O1) {
    int b = blockIdx.x;
    int wave = threadIdx.x >> 5, lane = threadIdx.x & 31;
    int i = blockIdx.y * 8 + wave;
    const float* u = U + ((size_t)b * HH + (i >> 7)) * DD;
    const float* w = Wv + (size_t)i * DD;
    float p = 0.f;
    for (int d = lane; d < DD; d += 32) p += u[d] * w[d];
    for (int off = 16; off > 0; off >>= 1) p += __shfl_xor(p, off, 32);
    if (lane == 0) O1[b * DD + i] = Qp[b * DD + i] + bv[i] + p;
}

// ---------------- out[b,c] = O2[b]·Wfc[c,:] + bfc[c]
__global__ void k_fc(const float* __restrict__ O2, const float* __restrict__ Wfc,
                     const float* __restrict__ bfc, float* __restrict__ out) {
    int b = blockIdx.x, c = blockIdx.y;
    const float* o = O2 + (size_t)b * DD;
    const float* w = Wfc + (size_t)c * DD;
    float p = 0.f;
    for (int d = threadIdx.x; d < DD; d += 32) p += o[d] * w[d];
    for (int off = 16; off > 0; off >>= 1) p += __shfl_xor(p, off, 32);
    if (threadIdx.x == 0) out[b * 2 + c] = p + bfc[c];
}

extern "C" void kernel_launch(void* const* d_in, const int* in_sizes, int n_in,
                              void* d_out, int out_size, void* d_ws, size_t ws_size,
                              hipStream_t stream) {
    const float* inputs    = (const float*)d_in[0];
    const float* W_enc     = (const float*)d_in[1];
    // d_in[2] b_enc: monotonic shift under sigmoid -> no effect on argmax
    const float* cls_token = (const float*)d_in[3];
    const float* conv_w    = (const float*)d_in[4];
    const float* conv_b    = (const float*)d_in[5];
    const float* Wq        = (const float*)d_in[6];
    const float* bq        = (const float*)d_in[7];
    const float* Wk        = (const float*)d_in[8];
    // d_in[9] bk: per-(b,h) constant on logits -> softmax-invariant, dropped
    const float* Wv        = (const float*)d_in[10];
    const float* bv        = (const float*)d_in[11];
    const float* Wo        = (const float*)d_in[12];
    const float* bo        = (const float*)d_in[13];
    const float* Wfc       = (const float*)d_in[14];
    const float* bfc       = (const float*)d_in[15];
    float* out = (float*)d_out;

    float* ws   = (float*)d_ws;
    float* X    = ws;                                   // B*4097*1024
    float* Q    = X    + (size_t)BB * NX * DD;          // B*1024
    float* Qp   = Q    + (size_t)BB * DD;               // B*1024
    float* G    = Qp   + (size_t)BB * DD;               // B*16*1024 (h>=8 zero)
    float* Lraw = G    + (size_t)BB * 16 * DD;          // B*4097*16
    float* Apd  = Lraw + (size_t)BB * NX * 16;          // B*16*4100 (zero-padded)
    float* U    = Apd  + (size_t)BB * 16 * APAD;        // B*8*1024
    float* O1   = U    + (size_t)BB * HH * DD;          // B*1024
    float* O2   = O1   + (size_t)BB * DD;               // B*1024
    int* idxbuf = (int*)(O2 + (size_t)BB * DD);         // B

    hipMemsetAsync(G,   0, (size_t)BB * 16 * DD   * sizeof(float), stream);
    hipMemsetAsync(Apd, 0, (size_t)BB * 16 * APAD * sizeof(float), stream);

    k_argmax<<<BB, 256, 0, stream>>>(inputs, W_enc, Q, idxbuf);
    k_cls   <<<BB, 256, 0, stream>>>(cls_token, X);
    k_conv  <<<dim3(BB, 64, DD / DC), 256, 0, stream>>>(inputs, Q, conv_w, conv_b, X);
    k_gemv_rowdot<<<dim3(BB, DD / 8), 256, 0, stream>>>(Q, Wq, bq, Qp, 0);
    k_g     <<<dim3(BB, HH, DD / 256), 256, 0, stream>>>(Qp, Wk, G);
    k_logits<<<dim3(BB, 65), 128, 0, stream>>>(X, G, Lraw);
    k_softmax<<<dim3(BB, HH), 256, 0, stream>>>(Lraw, Apd);
    k_u     <<<dim3(BB, 8), 256, 0, stream>>>(X, Apd, U);
    k_o     <<<dim3(BB, DD / 8), 256, 0, stream>>>(U, Wv, Qp, bv, O1);
    k_gemv_rowdot<<<dim3(BB, DD / 8), 256, 0, stream>>>(O1, Wo, bo, O2, 1);
    k_fc    <<<dim3(BB, 2), 32, 0, stream>>>(O2, Wfc, bfc, out);
}